// selfattention_9423158247594
// MI455X (gfx1250) — compile-verified
//
#include <hip/hip_runtime.h>
#include <stdint.h>

typedef __attribute__((ext_vector_type(2))) float v2f;
typedef __attribute__((ext_vector_type(8))) float v8f;

// D = A(16x4,f32) * B(4x16,f32) + C(16x16,f32)
#define WMMA_F32(a, b, c) \
  __builtin_amdgcn_wmma_f32_16x16x4_f32(false, (a), false, (b), (short)0, (c), false, false)

constexpr int Bsz = 2, H = 16, S = 2048, D = 64;
constexpr int MT  = 16;    // query rows per workgroup
constexpr int PAD = 2052;  // row stride in dwords: (2052%64)==4 -> conflict-free banks

__global__ __launch_bounds__(256)
void attn_fused(const float* __restrict__ q, const float* __restrict__ kmat,
                const float* __restrict__ vmat, const uint8_t* __restrict__ mask,
                const int* __restrict__ dk,
                float* __restrict__ out_res, float* __restrict__ out_attn) {
  __shared__ float sS[MT * PAD];   // full 16 x 2048 masked-score slab (~128 KB)
  __shared__ float sR[MT * D];     // cross-wave reduction for results tile
  __shared__ float sSum[MT];       // per-row 1/sum(exp)

  const int tid   = threadIdx.x;
  const int lane  = tid & 31;
  const int wv    = tid >> 5;      // wave 0..7
  const int mrow  = lane & 15;     // row/col-in-tile for operand loads
  const int khalf = lane >> 4;     // selects K pair {0,1} vs {2,3}

  const int bh    = blockIdx.x >> 7;      // S/MT = 128 m-tiles per (b,h)
  const int mtile = blockIdx.x & 127;
  const int b     = bh >> 4;              // H = 16
  const int m0    = mtile * MT;

  const float scale = rsqrtf((float)dk[0]);   // 1/sqrt(d_k)

  for (int i = tid; i < MT * D; i += 256) sR[i] = 0.0f;
  __syncthreads();

  const float*   Qb = q    + ((size_t)bh * S + m0) * D;
  const float*   Kb = kmat + (size_t)bh * S * D;
  const float*   Vb = vmat + (size_t)bh * S * D;
  const uint8_t* Mb = mask + (size_t)b * S * S;   // mask broadcast over H

  // ---- Q tile as 16 A-operands (K-steps of 4 over D=64), kept in VGPRs ----
  v2f qa[16];
#pragma unroll
  for (int kk = 0; kk < 16; ++kk)
    qa[kk] = *(const v2f*)(Qb + mrow * D + 4 * kk + 2 * khalf);

  // results accumulators: 16x64 as 4 n-tiles of 16x16 (4 independent chains)
  v8f racc[4];
#pragma unroll
  for (int nt = 0; nt < 4; ++nt) racc[nt] = (v8f){};

  // ---- per-wave: 16 contiguous t-tiles of the score row-slab ----
  for (int it = 0; it < 16; ++it) {
    const int t0 = (wv * 16 + it) * 16;
    const float* Kt = Kb + (size_t)t0 * D;
    if (it + 1 < 16)  // speculative prefetch of next K tile (global_prefetch_b8)
      __builtin_prefetch(Kt + 16 * D + mrow * D, 0, 0);

    // GEMM1: score tile = Q(16x64) * K(16x64)^T
    // Two independent accumulator chains (even/odd k-steps) to double
    // matrix-pipe ILP instead of one 16-deep dependent WMMA chain.
    v8f c0 = (v8f){};
    v8f c1 = (v8f){};
#pragma unroll
    for (int kk = 0; kk < 16; kk += 2) {
      v2f kb0 = *(const v2f*)(Kt + mrow * D + 4 * kk + 2 * khalf);
      v2f kb1 = *(const v2f*)(Kt + mrow * D + 4 * (kk + 1) + 2 * khalf);
      c0 = WMMA_F32(qa[kk],     kb0, c0);
      c1 = WMMA_F32(qa[kk + 1], kb1, c1);
    }

    // merge chains + scale + mask, stash in LDS slab
    // (D layout: M = r + 8*khalf, N = mrow)
    const int col = t0 + mrow;
#pragma unroll
    for (int r = 0; r < 8; ++r) {
      const int row = r + 8 * khalf;
      float sc  = (c0[r] + c1[r]) * scale;
      float val = Mb[(size_t)(m0 + row) * S + col] ? -1.0e9f : sc;
      sS[row * PAD + col] = val;
    }

    // GEMM2 partial: racc += S_tile(16x16) * V[t0:t0+16, 0:64]
    // (LDS round-trip converts the tile from D layout to A layout)
#pragma unroll
    for (int kk = 0; kk < 4; ++kk) {
      const int kg = t0 + 4 * kk + 2 * khalf;           // per-lane K row
      v2f a = *(const v2f*)&sS[mrow * PAD + kg];        // A[M=mrow][kg..kg+1]
      const float* Vr = Vb + (size_t)kg * D;            // rows kg, kg+1
#pragma unroll
      for (int nt = 0; nt < 4; ++nt) {
        v2f bb;
        bb.x = Vr[nt * 16 + mrow];
        bb.y = Vr[D + nt * 16 + mrow];
        racc[nt] = WMMA_F32(a, bb, racc[nt]);
      }
    }
  }

  // ---- cross-wave reduction of results tile via LDS float atomics ----
#pragma unroll
  for (int nt = 0; nt < 4; ++nt)
#pragma unroll
    for (int r = 0; r < 8; ++r) {
      const int row = r + 8 * khalf;
      atomicAdd(&sR[row * D + nt * 16 + mrow], racc[nt][r]);
    }
  __syncthreads();   // sS slab complete + sR complete

  // ---- softmax over each 2048-wide row: 16 teams of 16 lanes ----
  const int srow = tid >> 4;
  const int tl   = tid & 15;
  float mx = -3.402823466e38f;
  for (int j = tl; j < S; j += 16) mx = fmaxf(mx, sS[srow * PAD + j]);
#pragma unroll
  for (int m = 8; m >= 1; m >>= 1) mx = fmaxf(mx, __shfl_xor(mx, m, 32));
  float sum = 0.0f;
  for (int j = tl; j < S; j += 16) {
    float e = __expf(sS[srow * PAD + j] - mx);
    sS[srow * PAD + j] = e;            // overwrite slab with exp values
    sum += e;
  }
#pragma unroll
  for (int m = 8; m >= 1; m >>= 1) sum += __shfl_xor(sum, m, 32);
  if (tl == 0) sSum[srow] = 1.0f / sum;
  __syncthreads();

  // ---- stream attention out (write-once 537MB total -> non-temporal) ----
  float* attnB = out_attn + ((size_t)bh * S + m0) * S;
  for (int jj = 0; jj < (MT * S) / 256; ++jj) {
    const int idx = tid + jj * 256;
    const int row = idx >> 11;         // /2048
    const int col = idx & (S - 1);
    float val = sS[row * PAD + col] * sSum[row];
    __builtin_nontemporal_store(val, attnB + (size_t)row * S + col);
  }

  // ---- results tile out ----
  float* resB = out_res + ((size_t)bh * S + m0) * D;
#pragma unroll
  for (int i = 0; i < 4; ++i) {
    const int idx = tid + i * 256;     // row-major 16x64 block, contiguous
    __builtin_nontemporal_store(sR[idx], resB + idx);
  }
}

extern "C" void kernel_launch(void* const* d_in, const int* in_sizes, int n_in,
                              void* d_out, int out_size, void* d_ws, size_t ws_size,
                              hipStream_t stream) {
  const float*   q    = (const float*)d_in[0];
  const float*   k    = (const float*)d_in[1];
  const float*   v    = (const float*)d_in[2];
  const uint8_t* mask = (const uint8_t*)d_in[3];
  const int*     dk   = (const int*)d_in[4];

  float* out_res  = (float*)d_out;                         // [B,H,S,D]
  float* out_attn = out_res + (size_t)Bsz * H * S * D;     // [B,H,S,S]

  dim3 grid(Bsz * H * (S / MT));   // 4096 workgroups
  attn_fused<<<grid, 256, 0, stream>>>(q, k, v, mask, dk, out_res, out_attn);
}